// GDTM_16569983828458
// MI455X (gfx1250) — compile-verified
//
#include <hip/hip_runtime.h>
#include <hip/hip_bf16.h>
#include <math.h>

// Problem constants from the reference
#define T_STEPS 512
#define V_DIM   40000
#define K_DIM   50
#define H_DIM   200
#define G_DIM   800   // 4*H

typedef __attribute__((ext_vector_type(2))) float v2f;
typedef __attribute__((ext_vector_type(8))) float v8f;

#if __has_builtin(__builtin_amdgcn_global_load_async_to_lds_b32) && \
    __has_builtin(__builtin_amdgcn_s_wait_asynccnt)
#define HAVE_ASYNC_LDS 1
typedef __attribute__((address_space(1))) int* gint_t;   // global src
typedef __attribute__((address_space(3))) int* sint_t;   // LDS dst
#else
#define HAVE_ASYNC_LDS 0
#endif

__device__ __forceinline__ float sigmoidf_(float x) {
    return 1.0f / (1.0f + __expf(-x));
}

// ---------------------------------------------------------------------------
// WMMA fp32 GEMM:  C[M,N] = A[M,K] * B^T + bias0 (+ bias1)
//   A row-major (row stride sA); B stored as Bw[n*sB + k] (B^T row-major).
//   One wave computes a 16 x (16*NT) strip: the A fragment is reused across
//   NT B fragments / NT independent accumulators. Software-pipelined:
//   fragments for K-step k+4 are loaded before the WMMAs for step k issue,
//   so s_wait_loadcnt covers loads issued one full WMMA-group earlier.
//   Out-of-range columns are CLAMPED on load (a WMMA output column depends
//   only on its own B column) and guarded only at store — no exec branches.
// A-fragment (32-bit 16x4): lane&15 = M, hi = lane>>4:
//   a.x = A[m][k0+2*hi], a.y = A[m][k0+2*hi+1]
// B-fragment (4x16): lane&15 = N, v0 = K rows {k0+2*hi}, v1 = {k0+2*hi+1}
// C/D: vgpr v holds (M = v + 8*hi, N = lane&15)
// ---------------------------------------------------------------------------
template <int NT>
__global__ void wmma_gemm_bt_f32(const float* __restrict__ A,
                                 const float* __restrict__ Bw,
                                 const float* __restrict__ bias0,
                                 const float* __restrict__ bias1,
                                 float* __restrict__ C,
                                 int N, int Ktot, int sA, int sB, int ldc) {
    const int lane = threadIdx.x & 31;
    const int lo   = lane & 15;
    const int hi   = lane >> 4;
    const int m    = blockIdx.x * 16 + lo;

    v8f          c[NT];
    const float* brow[NT];
    int          ncol[NT];

    #pragma unroll
    for (int u = 0; u < NT; ++u) {
        int n   = blockIdx.y * (16 * NT) + u * 16 + lo;
        ncol[u] = n;
        int nc  = (n < N) ? n : (N - 1);          // clamp: branch-free loads
        float bv = bias0[nc];
        if (bias1) bv += bias1[nc];
        #pragma unroll
        for (int i = 0; i < 8; ++i) c[u][i] = bv;
        brow[u] = Bw + (size_t)nc * sB;
    }

    const float* arow = A + (size_t)m * sA;

    // --- software-pipelined K loop (Ktot is a multiple of 4, >= 8) ---
    v2f a_cur, b_cur[NT];
    {
        const int k = 2 * hi;
        a_cur = *reinterpret_cast<const v2f*>(arow + k);
        #pragma unroll
        for (int u = 0; u < NT; ++u)
            b_cur[u] = *reinterpret_cast<const v2f*>(brow[u] + k);
    }

    for (int k0 = 0; k0 + 4 < Ktot; k0 += 4) {
        const int kn = k0 + 4 + 2 * hi;
        v2f a_nxt = *reinterpret_cast<const v2f*>(arow + kn);
        v2f b_nxt[NT];
        #pragma unroll
        for (int u = 0; u < NT; ++u)
            b_nxt[u] = *reinterpret_cast<const v2f*>(brow[u] + kn);

        #pragma unroll
        for (int u = 0; u < NT; ++u) {
            // 8 args: (neg_a, A, neg_b, B, c_mod, C, reuse_a, reuse_b)
            c[u] = __builtin_amdgcn_wmma_f32_16x16x4_f32(
                       false, a_cur, false, b_cur[u], (short)0, c[u],
                       false, false);
        }
        a_cur = a_nxt;
        #pragma unroll
        for (int u = 0; u < NT; ++u) b_cur[u] = b_nxt[u];
    }
    #pragma unroll
    for (int u = 0; u < NT; ++u) {
        c[u] = __builtin_amdgcn_wmma_f32_16x16x4_f32(
                   false, a_cur, false, b_cur[u], (short)0, c[u],
                   false, false);
    }

    const int mbase = blockIdx.x * 16 + 8 * hi;
    #pragma unroll
    for (int u = 0; u < NT; ++u) {
        if (ncol[u] < N) {
            #pragma unroll
            for (int v = 0; v < 8; ++v) {
                C[(size_t)(mbase + v) * ldc + ncol[u]] = c[u][v];
            }
        }
    }
}

// ---------------------------------------------------------------------------
// Transpose w_hh [4H, H] -> whhT [H, 4H] so the scan kernel's per-step loads
// are wave-coalesced (thread i reads whhT[j*800 + i]).
// ---------------------------------------------------------------------------
__global__ void transpose_whh(const float* __restrict__ whh,
                              float* __restrict__ whhT) {
    int idx = blockIdx.x * blockDim.x + threadIdx.x;   // over 800*200
    if (idx < G_DIM * H_DIM) {
        int i = idx / H_DIM;   // gate row 0..799
        int j = idx % H_DIM;   // hidden col 0..199
        whhT[j * G_DIM + i] = whh[idx];
    }
}

// ---------------------------------------------------------------------------
// Sequential LSTM scan for one layer. Single workgroup of 800 threads
// (25 wave32). h, c, gates in LDS; w_hh^T streamed coalesced from L2
// (640 KB, stays hot). h broadcast reads are ds_load_b128 (float4).
// The next timestep's xw row is staged into an LDS double-buffer with
// global_load_async_to_lds_b32 (ASYNCcnt) so its fetch overlaps the GEMV;
// each thread reads back only the word it loaded, so a wave-local
// s_wait_asynccnt suffices (no extra barrier). Falls back to
// global_prefetch if the async builtins are unavailable.
// Gate order (torch): i, f, g, o along the 4H axis.
// ---------------------------------------------------------------------------
__global__ void lstm_scan(const float* __restrict__ xw,    // [T, 800]
                          const float* __restrict__ whhT,  // [200, 800]
                          float* __restrict__ hseq) {      // [T, 200]
    __shared__ __align__(16) float hs[H_DIM];
    __shared__ float cs[H_DIM];
    __shared__ float gates[G_DIM];
#if HAVE_ASYNC_LDS
    __shared__ float xwbuf[2][G_DIM];
#endif

    const int i = threadIdx.x;            // 0..799
    if (i < H_DIM) { hs[i] = 0.0f; cs[i] = 0.0f; }
    __syncthreads();

    const float* wcol = whhT + i;

#if HAVE_ASYNC_LDS
    // prologue: stage xw[0] asynchronously
    __builtin_amdgcn_global_load_async_to_lds_b32(
        (gint_t)(xw + i), (sint_t)&xwbuf[0][i], 0, 0);
#endif

    for (int t = 0; t < T_STEPS; ++t) {
#if HAVE_ASYNC_LDS
        __builtin_amdgcn_s_wait_asynccnt(0);          // xw[t] staged
        float acc = xwbuf[t & 1][i];
        if (t + 1 < T_STEPS) {                        // stage xw[t+1]
            __builtin_amdgcn_global_load_async_to_lds_b32(
                (gint_t)(xw + (t + 1) * G_DIM + i),
                (sint_t)&xwbuf[(t + 1) & 1][i], 0, 0);
        }
#else
        float acc = xw[t * G_DIM + i];
        if (t + 1 < T_STEPS) {
            __builtin_prefetch(&xw[(t + 1) * G_DIM + i], 0, 1);
        }
#endif

        const float4* hs4 = reinterpret_cast<const float4*>(hs);
        #pragma unroll 2
        for (int j4 = 0; j4 < H_DIM / 4; ++j4) {
            float4 h4 = hs4[j4];                      // LDS broadcast, b128
            const float* w = wcol + (size_t)(j4 * 4) * G_DIM;
            acc += h4.x * w[0 * G_DIM];
            acc += h4.y * w[1 * G_DIM];
            acc += h4.z * w[2 * G_DIM];
            acc += h4.w * w[3 * G_DIM];
        }
        gates[i] = acc;
        __syncthreads();

        if (i < H_DIM) {
            float ig = sigmoidf_(gates[i]);
            float fg = sigmoidf_(gates[i + 1 * H_DIM]);
            float gg = tanhf    (gates[i + 2 * H_DIM]);
            float og = sigmoidf_(gates[i + 3 * H_DIM]);
            float cn = fg * cs[i] + ig * gg;
            float hn = og * tanhf(cn);
            cs[i] = cn;
            hs[i] = hn;
            hseq[t * H_DIM + i] = hn;
        }
        __syncthreads();
    }
}

// ---------------------------------------------------------------------------
// Sequential eta chain. Single workgroup of 128 threads.
//   inp_t = [h3_t (200), eta_prev (50)]  (250)
//   mu = inp_t @ w_mu^T + b_mu ; ls = inp_t @ w_ls^T + b_ls ; eta = mu
//   KL accumulated serially (deterministic order) by thread 0.
// Output: etas[512,50] then kl scalar at out[25600].
// ---------------------------------------------------------------------------
__global__ void eta_scan(const float* __restrict__ h3,
                         const float* __restrict__ wmu, const float* __restrict__ bmu,
                         const float* __restrict__ wls, const float* __restrict__ bls,
                         float* __restrict__ out) {
    __shared__ float inp[256];     // 0..199 h, 200..249 eta_prev
    __shared__ float muv[64];
    __shared__ float lsv[64];
    __shared__ float klp[64];

    const int tid = threadIdx.x;   // 0..127
    float kl_acc = 0.0f;           // meaningful on thread 0 only

    if (tid < 56) inp[H_DIM + tid] = 0.0f;   // eta_prev = 0 at t=0 (+pad)
    if (tid < 64) klp[tid] = 0.0f;
    __syncthreads();

    const float ood0 = 1.0f / (1.0f + 1e-6f);
    const float ood  = 1.0f / (0.005f + 1e-6f);
    const float logd = logf(0.005f);

    for (int t = 0; t < T_STEPS; ++t) {
        // load h3[t] into LDS (200 values over 128 threads)
        inp[tid] = h3[t * H_DIM + tid];
        if (tid < H_DIM - 128) inp[128 + tid] = h3[t * H_DIM + 128 + tid];
        __syncthreads();

        if (tid < K_DIM) {
            float a = bmu[tid];
            const float* w = wmu + tid * (H_DIM + K_DIM);
            for (int j = 0; j < H_DIM + K_DIM; ++j) a += inp[j] * w[j];
            muv[tid] = a;
        } else if (tid >= 64 && tid < 64 + K_DIM) {
            const int k = tid - 64;
            float a = bls[k];
            const float* w = wls + k * (H_DIM + K_DIM);
            for (int j = 0; j < H_DIM + K_DIM; ++j) a += inp[j] * w[j];
            lsv[k] = a;
        }
        __syncthreads();

        if (tid < K_DIM) {
            float mu = muv[tid];
            float ls = lsv[tid];
            float ep = inp[H_DIM + tid];
            float term;
            if (t == 0) {
                term = 0.5f * ((__expf(ls) + mu * mu) * ood0 - 1.0f - ls);
            } else {
                float d = mu - ep;
                term = 0.5f * ((__expf(ls) + d * d) * ood - 1.0f + logd - ls);
            }
            klp[tid] = term;
            out[t * K_DIM + tid] = mu;     // eta = mu (eval mode)
            inp[H_DIM + tid] = mu;         // eta_prev for next step
        }
        __syncthreads();

        if (tid == 0) {
            float s = 0.0f;
            for (int k = 0; k < K_DIM; ++k) s += klp[k];
            kl_acc += s;
        }
        __syncthreads();
    }

    if (tid == 0) out[T_STEPS * K_DIM] = kl_acc;
}

// ---------------------------------------------------------------------------
// Launch: map GEMM -> per layer (transpose w_hh, gate GEMM, scan) -> eta chain
// ---------------------------------------------------------------------------
extern "C" void kernel_launch(void* const* d_in, const int* in_sizes, int n_in,
                              void* d_out, int out_size, void* d_ws, size_t ws_size,
                              hipStream_t stream) {
    const float* rnn_inp = (const float*)d_in[0];
    const float* w_map   = (const float*)d_in[1];
    const float* b_map   = (const float*)d_in[2];
    const float* w_mu    = (const float*)d_in[3];
    const float* b_mu    = (const float*)d_in[4];
    const float* w_ls    = (const float*)d_in[5];
    const float* b_ls    = (const float*)d_in[6];
    const float* w_ih[3] = { (const float*)d_in[7],  (const float*)d_in[11], (const float*)d_in[15] };
    const float* w_hh[3] = { (const float*)d_in[8],  (const float*)d_in[12], (const float*)d_in[16] };
    const float* b_ih[3] = { (const float*)d_in[9],  (const float*)d_in[13], (const float*)d_in[17] };
    const float* b_hh[3] = { (const float*)d_in[10], (const float*)d_in[14], (const float*)d_in[18] };

    float* ws = (float*)d_ws;
    float* out0 = ws;                          // [512, 200]
    float* xw   = out0 + T_STEPS * H_DIM;      // [512, 800]
    float* hA   = xw   + T_STEPS * G_DIM;      // [512, 200]
    float* hB   = hA   + T_STEPS * H_DIM;      // [512, 200]
    float* whhT = hB   + T_STEPS * H_DIM;      // [200, 800]

    constexpr int NT = 4;                      // 16x64 strip per wave

    // 1) out0 = rnn_inp @ w_map^T + b_map     (M=512, N=200, K=40000)
    {
        dim3 grid(T_STEPS / 16, (H_DIM + 16 * NT - 1) / (16 * NT));
        wmma_gemm_bt_f32<NT><<<grid, 32, 0, stream>>>(
            rnn_inp, w_map, b_map, nullptr, out0,
            H_DIM, V_DIM, V_DIM, V_DIM, H_DIM);
    }

    // 2) three LSTM layers
    const float* x_l = out0;
    float* hbuf[3] = { hA, hB, hA };
    for (int l = 0; l < 3; ++l) {
        transpose_whh<<<(G_DIM * H_DIM + 255) / 256, 256, 0, stream>>>(w_hh[l], whhT);

        // xw = x_l @ w_ih^T + (b_ih + b_hh)   (M=512, N=800, K=200)
        dim3 grid(T_STEPS / 16, (G_DIM + 16 * NT - 1) / (16 * NT));
        wmma_gemm_bt_f32<NT><<<grid, 32, 0, stream>>>(
            x_l, w_ih[l], b_ih[l], b_hh[l], xw,
            G_DIM, H_DIM, H_DIM, H_DIM, G_DIM);

        lstm_scan<<<1, G_DIM, 0, stream>>>(xw, whhT, hbuf[l]);
        x_l = hbuf[l];
    }

    // 3) eta chain + KL
    eta_scan<<<1, 128, 0, stream>>>(hbuf[2], w_mu, b_mu, w_ls, b_ls, (float*)d_out);
}